// GatedSwitchesGNN_90108413870585
// MI455X (gfx1250) — compile-verified
//
#include <hip/hip_runtime.h>
#include <hip/hip_bf16.h>

#define B_  4
#define V_  256
#define H_  128
#define BV  (B_*V_)          // 1024
#define NE  (B_*V_*V_)       // 262144
#define EPSF 1e-5f

#define LDS_STRIDE 132       // 128 + 4 floats pad -> avoids 16-way bank conflicts on A-frag reads
#define ROWS 64              // e rows per block in the big passes
#define NSLOT 64             // contention-splitting slots for e-channel stats atomics

typedef __attribute__((ext_vector_type(2))) float v2f;
typedef __attribute__((ext_vector_type(8))) float v8f;
typedef int vec4i __attribute__((vector_size(16)));   // matches builtin param type

#define AS1 __attribute__((address_space(1)))
#define AS3 __attribute__((address_space(3)))

#if __has_builtin(__builtin_amdgcn_global_load_async_to_lds_b128)
#define USE_ASYNC_LDS 1
#endif

__device__ __forceinline__ void wait_async_and_barrier() {
#if __has_builtin(__builtin_amdgcn_s_wait_asynccnt)
  __builtin_amdgcn_s_wait_asynccnt(0);
#else
  asm volatile("s_wait_asynccnt 0" ::: "memory");
#endif
  __syncthreads();
}

__device__ __forceinline__ v8f wmma_f32(v2f a, v2f b, v8f c) {
  // D = A(16x4 f32) x B(4x16 f32) + C(16x16 f32)
  return __builtin_amdgcn_wmma_f32_16x16x4_f32(false, a, false, b, (short)0, c,
                                               false, false);
}

// Load a ROWS x 128 fp32 tile of `src` (row base r0) into LDS (stride LDS_STRIDE).
// Uses the gfx1250 async global->LDS path when available (no VGPR round-trip).
__device__ __forceinline__ void load_tile64(const float* __restrict__ src,
                                            int r0, float* lds, int tid) {
#ifdef USE_ASYNC_LDS
#pragma unroll
  for (int it = 0; it < 8; ++it) {
    int q = tid + 256 * it;
    int r = q >> 5, c4 = q & 31;
    const float* gp = src + (size_t)(r0 + r) * H_ + c4 * 4;
    float*       lp = &lds[r * LDS_STRIDE + c4 * 4];
    __builtin_amdgcn_global_load_async_to_lds_b128(
        (AS1 vec4i*)gp, (AS3 vec4i*)lp, 0, 0);
  }
  wait_async_and_barrier();
#else
#pragma unroll
  for (int it = 0; it < 8; ++it) {
    int q = tid + 256 * it;
    int r = q >> 5, c4 = q & 31;
    float4 t = *(const float4*)(src + (size_t)(r0 + r) * H_ + c4 * 4);
    *(float4*)(&lds[r * LDS_STRIDE + c4 * 4]) = t;
  }
  __syncthreads();
#endif
}

// ---------------------------------------------------------------------------
// Kernel 1: Uh/Vh/Ah/Bh = h @ W.T + b   (grid.y selects which projection)
// ---------------------------------------------------------------------------
__global__ __launch_bounds__(256) void k_proj(
    const float* __restrict__ h,
    const float* __restrict__ WU, const float* __restrict__ bU,
    const float* __restrict__ WV, const float* __restrict__ bV,
    const float* __restrict__ WA, const float* __restrict__ bA,
    const float* __restrict__ WB, const float* __restrict__ bB,
    float* __restrict__ Uh, float* __restrict__ Vh,
    float* __restrict__ Ah, float* __restrict__ Bh)
{
  __shared__ float lds[16 * LDS_STRIDE];
  const int tid = threadIdx.x;
  const int m0  = blockIdx.x * 16;

  const float* W; const float* bias; float* out;
  switch (blockIdx.y) {
    case 0:  W = WU; bias = bU; out = Uh; break;
    case 1:  W = WV; bias = bV; out = Vh; break;
    case 2:  W = WA; bias = bA; out = Ah; break;
    default: W = WB; bias = bB; out = Bh; break;
  }

  // cooperative load: 16 x 128 fp32 tile of h
#pragma unroll
  for (int it = 0; it < 2; ++it) {
    int q  = tid + 256 * it;          // 0..511 float4s
    int r  = q >> 5, c4 = q & 31;
    float4 t = *(const float4*)(h + (size_t)(m0 + r) * H_ + c4 * 4);
    *(float4*)(&lds[r * LDS_STRIDE + c4 * 4]) = t;
  }
  __syncthreads();

  const int lane = tid & 31, wv = tid >> 5;   // 8 waves -> 8 N-tiles
  const int ln = lane & 15, hi = lane >> 4;
  const int n  = wv * 16 + ln;

  v8f acc = {};
  const float* Wr = W + (size_t)n * H_;       // B[k][n] = W[n][k]
#pragma unroll 8
  for (int kk = 0; kk < H_; kk += 4) {
    v2f bfrag = *(const v2f*)(Wr + kk + 2 * hi);
    v2f afrag = *(const v2f*)(&lds[ln * LDS_STRIDE + kk + 2 * hi]);
    acc = wmma_f32(afrag, bfrag, acc);
  }

  float bn = bias[n];
#pragma unroll
  for (int vv = 0; vv < 8; ++vv) {
    out[(size_t)(m0 + vv + 8 * hi) * H_ + n] = acc[vv] + bn;
  }
}

// ---------------------------------------------------------------------------
// Kernel 2: pass 1 over e. Recompute e_new in registers, produce:
//   agg[b,i,:] += sum_j ((S?0:sigmoid(e_new)) + (A?0:1)) * Vh[b,j,:]
//   per-channel sum / sumsq of e_new (sliced atomics)
// ---------------------------------------------------------------------------
__global__ __launch_bounds__(256) void k_pass1(
    const float* __restrict__ e,
    const unsigned char* __restrict__ Am, const unsigned char* __restrict__ Sm,
    const float* __restrict__ WC, const float* __restrict__ bC,
    const float* __restrict__ Ah, const float* __restrict__ Bh,
    const float* __restrict__ Vh,
    float* __restrict__ agg, float* __restrict__ esum, float* __restrict__ esq)
{
  __shared__ float lds[ROWS * LDS_STRIDE];
  const int tid = threadIdx.x;
  const int r0  = blockIdx.x * ROWS;          // flattened (b,i,j) row
  const int b   = r0 >> 16;
  const int i   = (r0 >> 8) & 255;
  const int j0  = r0 & 255;                   // 64-aligned -> single i per block

  load_tile64(e, r0, lds, tid);

  const int lane = tid & 31, wv = tid >> 5;
  const int ln = lane & 15, hi = lane >> 4;
  const int n  = wv * 16 + ln;

  v8f acc[4] = {};
  const float* Wr = WC + (size_t)n * H_;
#pragma unroll 8
  for (int kk = 0; kk < H_; kk += 4) {
    v2f bfrag = *(const v2f*)(Wr + kk + 2 * hi);
    const float* ab = &lds[ln * LDS_STRIDE + kk + 2 * hi];
#pragma unroll
    for (int mt = 0; mt < 4; ++mt) {
      v2f afrag = *(const v2f*)(ab + mt * 16 * LDS_STRIDE);
      acc[mt] = wmma_f32(afrag, bfrag, acc[mt]);
    }
  }

  const float bCn = bC[n];
  const float Bhn = Bh[((size_t)b * V_ + i) * H_ + n];
  const size_t mbase = (size_t)b * V_ * V_ + (size_t)i * V_;

  float ssum = 0.f, ssq = 0.f, ap = 0.f;
#pragma unroll
  for (int mt = 0; mt < 4; ++mt) {
#pragma unroll
    for (int vv = 0; vv < 8; ++vv) {
      int j = j0 + mt * 16 + vv + 8 * hi;     // D layout: VGPR vv -> M=vv (+8 for hi half)
      float en = acc[mt][vv] + bCn + Ah[((size_t)b * V_ + j) * H_ + n] + Bhn;
      float g  = 1.0f / (1.0f + __expf(-en));
      float coef = (Sm[mbase + j] ? 0.0f : g) + (Am[mbase + j] ? 0.0f : 1.0f);
      ap   += coef * Vh[((size_t)b * V_ + j) * H_ + n];
      ssum += en;
      ssq  += en * en;
    }
  }
  // lanes L and L+16 hold the same channel n -> fold, then one atomic per n
  ap   += __shfl_down(ap, 16);
  ssum += __shfl_down(ssum, 16);
  ssq  += __shfl_down(ssq, 16);
  if (hi == 0) {
    atomicAdd(&agg[((size_t)b * V_ + i) * H_ + n], ap);
    int slot = blockIdx.x & (NSLOT - 1);
    atomicAdd(&esum[slot * H_ + n], ssum);
    atomicAdd(&esq [slot * H_ + n], ssq);
  }
}

// ---------------------------------------------------------------------------
// Kernel 3: finalize stats, emit h_out (single block)
// ---------------------------------------------------------------------------
__global__ __launch_bounds__(256) void k_finalize(
    const float* __restrict__ h_in,
    const float* __restrict__ Uh, const float* __restrict__ agg,
    const float* __restrict__ esum, const float* __restrict__ esq,
    const float* __restrict__ gamma_h, const float* __restrict__ beta_h,
    float* __restrict__ me_out, float* __restrict__ rse_out,
    float* __restrict__ h_out)
{
  __shared__ float s_mh[H_], s_rs[H_];
  __shared__ float psum[256], psq[256];
  const int t = threadIdx.x;

  // e-channel statistics: reduce NSLOT partial slots
  if (t < H_) {
    float se = 0.f, sq = 0.f;
    for (int s = 0; s < NSLOT; ++s) { se += esum[s * H_ + t]; sq += esq[s * H_ + t]; }
    float me = se / (float)NE;
    float ve = sq / (float)NE - me * me;
    me_out[t]  = me;
    rse_out[t] = rsqrtf(ve + EPSF);
  }

  // h-channel statistics over 1024 rows of (Uh + agg)
  {
    int ch = t & 127, half = t >> 7;
    float s = 0.f, q = 0.f;
    for (int r = half * 512; r < half * 512 + 512; ++r) {
      float x = Uh[(size_t)r * H_ + ch] + agg[(size_t)r * H_ + ch];
      s += x; q += x * x;
    }
    psum[t] = s; psq[t] = q;
  }
  __syncthreads();
  if (t < H_) {
    float s = psum[t] + psum[t + 128];
    float q = psq[t]  + psq[t + 128];
    float m = s / (float)BV;
    float v = q / (float)BV - m * m;
    s_mh[t] = m;
    s_rs[t] = rsqrtf(v + EPSF);
  }
  __syncthreads();

  // h_out = h_in + relu(BN(Uh + agg))
  for (int it = 0; it < (BV * H_) / 256; ++it) {
    int idx = t + 256 * it;
    int c   = idx & 127;
    float x = Uh[idx] + agg[idx];
    float y = (x - s_mh[c]) * s_rs[c] * gamma_h[c] + beta_h[c];
    y = y > 0.f ? y : 0.f;
    h_out[idx] = h_in[idx] + y;
  }
}

// ---------------------------------------------------------------------------
// Kernel 4: pass 2 over e. Recompute e_new, normalize, e_out = e + relu(BN(e_new))
// (e tile in LDS serves as both GEMM A-matrix and the residual source)
// ---------------------------------------------------------------------------
__global__ __launch_bounds__(256) void k_pass2(
    const float* __restrict__ e,
    const float* __restrict__ WC, const float* __restrict__ bC,
    const float* __restrict__ Ah, const float* __restrict__ Bh,
    const float* __restrict__ meA, const float* __restrict__ rseA,
    const float* __restrict__ gamma_e, const float* __restrict__ beta_e,
    float* __restrict__ e_out)
{
  __shared__ float lds[ROWS * LDS_STRIDE];
  const int tid = threadIdx.x;
  const int r0  = blockIdx.x * ROWS;
  const int b   = r0 >> 16;
  const int i   = (r0 >> 8) & 255;
  const int j0  = r0 & 255;

  load_tile64(e, r0, lds, tid);

  const int lane = tid & 31, wv = tid >> 5;
  const int ln = lane & 15, hi = lane >> 4;
  const int n  = wv * 16 + ln;

  v8f acc[4] = {};
  const float* Wr = WC + (size_t)n * H_;
#pragma unroll 8
  for (int kk = 0; kk < H_; kk += 4) {
    v2f bfrag = *(const v2f*)(Wr + kk + 2 * hi);
    const float* ab = &lds[ln * LDS_STRIDE + kk + 2 * hi];
#pragma unroll
    for (int mt = 0; mt < 4; ++mt) {
      v2f afrag = *(const v2f*)(ab + mt * 16 * LDS_STRIDE);
      acc[mt] = wmma_f32(afrag, bfrag, acc[mt]);
    }
  }

  const float bCn = bC[n];
  const float Bhn = Bh[((size_t)b * V_ + i) * H_ + n];
  const float men = meA[n], rs = rseA[n], ga = gamma_e[n], be = beta_e[n];

#pragma unroll
  for (int mt = 0; mt < 4; ++mt) {
#pragma unroll
    for (int vv = 0; vv < 8; ++vv) {
      int rt = mt * 16 + vv + 8 * hi;         // row within tile
      int j  = j0 + rt;
      float en = acc[mt][vv] + bCn + Ah[((size_t)b * V_ + j) * H_ + n] + Bhn;
      float y  = (en - men) * rs * ga + be;
      y = y > 0.f ? y : 0.f;
      e_out[(size_t)(r0 + rt) * H_ + n] = lds[rt * LDS_STRIDE + n] + y;
    }
  }
}

// ---------------------------------------------------------------------------
extern "C" void kernel_launch(void* const* d_in, const int* in_sizes, int n_in,
                              void* d_out, int out_size, void* d_ws, size_t ws_size,
                              hipStream_t stream) {
  const float* h  = (const float*)d_in[0];
  const float* e  = (const float*)d_in[1];
  const unsigned char* Am = (const unsigned char*)d_in[2];
  const unsigned char* Sm = (const unsigned char*)d_in[3];
  const float* WU = (const float*)d_in[4];  const float* bU = (const float*)d_in[5];
  const float* WV = (const float*)d_in[6];  const float* bV = (const float*)d_in[7];
  const float* WA = (const float*)d_in[8];  const float* bA = (const float*)d_in[9];
  const float* WB = (const float*)d_in[10]; const float* bB = (const float*)d_in[11];
  const float* WC = (const float*)d_in[12]; const float* bC = (const float*)d_in[13];
  const float* gh = (const float*)d_in[14]; const float* bh = (const float*)d_in[15];
  const float* ge = (const float*)d_in[16]; const float* be = (const float*)d_in[17];

  float* ws   = (float*)d_ws;
  float* Uh   = ws;                    // BV*H
  float* Vh   = Uh  + (size_t)BV * H_;
  float* Ah   = Vh  + (size_t)BV * H_;
  float* Bh   = Ah  + (size_t)BV * H_;
  float* agg  = Bh  + (size_t)BV * H_; // BV*H   (zeroed)
  float* esum = agg + (size_t)BV * H_; // NSLOT*H (zeroed)
  float* esq  = esum + (size_t)NSLOT * H_;
  float* meA  = esq  + (size_t)NSLOT * H_;
  float* rseA = meA + H_;

  (void)hipMemsetAsync(agg, 0,
                       ((size_t)BV * H_ + 2 * (size_t)NSLOT * H_) * sizeof(float),
                       stream);

  float* h_out = (float*)d_out;
  float* e_out = h_out + (size_t)BV * H_;

  k_proj    <<<dim3(BV / 16, 4), 256, 0, stream>>>(h, WU, bU, WV, bV, WA, bA, WB, bB,
                                                   Uh, Vh, Ah, Bh);
  k_pass1   <<<NE / ROWS, 256, 0, stream>>>(e, Am, Sm, WC, bC, Ah, Bh, Vh,
                                            agg, esum, esq);
  k_finalize<<<1, 256, 0, stream>>>(h, Uh, agg, esum, esq, gh, bh, meA, rseA, h_out);
  k_pass2   <<<NE / ROWS, 256, 0, stream>>>(e, WC, bC, Ah, Bh, meA, rseA, ge, be,
                                            e_out);
}